// FeatSimLoss_64441689309909
// MI455X (gfx1250) — compile-verified
//
#include <hip/hip_runtime.h>

typedef float v2f __attribute__((ext_vector_type(2)));
typedef float v8f __attribute__((ext_vector_type(8)));

#define BATCH 2
#define CLOG  19
#define HH    128
#define WW    128
#define CF    320
#define HL    32
#define WL    32

// workspace layout (float offsets)
//   Gdiag  : [B][HL][3(dy)][3(dx)][WL]  = 2*32*3*3*32 = 18432 floats
//   prob   : [B][CLOG][H][W]            = 622592 floats
//   partial: [128][3]
#define GD_OFF   0
#define PROB_OFF 18432
#define PART_OFF (18432 + 622592)

// ---------------------------------------------------------------------------
// Kernel 1: low-res row-pair Gram via f32 WMMA.
//   G[b][y][dy+1][dx+1][x] = sum_c x_ema[b,c,y,x] * x_ema[b,c,y+dy,x+dx]
// One wave32 per (tile-pair, y, dy, b). 16x16 tile, K=320 in steps of 4.
// A layout (16x4 f32): lane<16 -> {K=kb,K=kb+1} rows M=lane ; lane>=16 -> K+2.
// D layout: lanes 0-15 N=lane M=vgpr(0..7); lanes 16-31 N=lane-16 M=vgpr+8.
// ---------------------------------------------------------------------------
__global__ __launch_bounds__(32)
void gram_wmma_kernel(const float* __restrict__ xe, float* __restrict__ gd) {
    const int tp  = blockIdx.x;          // 0..3 : tile pair
    const int txm = tp & 1;              // center-x tile
    const int txn = tp >> 1;             // neighbor-x tile
    const int y   = (int)(blockIdx.y) & 31;
    const int oy  = (int)(blockIdx.y >> 5) - 1;   // -1,0,1
    const int b   = blockIdx.z;
    const int y2  = y + oy;
    if (y2 < 0 || y2 >= HL) return;      // uniform across wave -> EXEC stays full

    const int lane = threadIdx.x;
    const int m    = lane & 15;
    const int kb   = (lane >> 4) << 1;   // 0 or 2

    const float* Xc = xe + (size_t)b * CF * HL * WL + (size_t)y  * WL + txm * 16 + m;
    const float* Xn = xe + (size_t)b * CF * HL * WL + (size_t)y2 * WL + txn * 16 + m;

    v8f acc = {0.f, 0.f, 0.f, 0.f, 0.f, 0.f, 0.f, 0.f};
    for (int c0 = 0; c0 < CF; c0 += 4) {
        v2f a, bb;
        a.x  = Xc[(size_t)(c0 + kb)     * (HL * WL)];
        a.y  = Xc[(size_t)(c0 + kb + 1) * (HL * WL)];
        bb.x = Xn[(size_t)(c0 + kb)     * (HL * WL)];
        bb.y = Xn[(size_t)(c0 + kb + 1) * (HL * WL)];
        acc = __builtin_amdgcn_wmma_f32_16x16x4_f32(
                  false, a, false, bb, (short)0, acc, false, false);
    }

    // scatter needed diagonals (dx in {-1,0,1}), indexed by center pixel gm
    const int n  = lane & 15;
    const int mh = (lane >> 4) << 3;
    const int gn = txn * 16 + n;
#pragma unroll
    for (int v = 0; v < 8; ++v) {
        int gm = txm * 16 + mh + v;
        int ox = gn - gm;
        if (ox >= -1 && ox <= 1) {
            gd[(((((size_t)b * HL + y) * 3 + (oy + 1)) * 3) + (ox + 1)) * WL + gm] = acc[v];
        }
    }
}

// ---------------------------------------------------------------------------
// Kernel 2: channel softmax (C=19) -> prob in workspace
// ---------------------------------------------------------------------------
__global__ __launch_bounds__(256)
void softmax_kernel(const float* __restrict__ lg, float* __restrict__ prob) {
    const int idx = blockIdx.x * 256 + threadIdx.x;      // over B*H*W = 32768
    const int b   = idx >> 14;
    const int hw  = idx & 16383;
    const float* lp = lg + (size_t)b * CLOG * HH * WW + hw;
    float vals[CLOG];
    float mx = -3.0e38f;
#pragma unroll
    for (int c = 0; c < CLOG; ++c) { vals[c] = lp[(size_t)c * HH * WW]; mx = fmaxf(mx, vals[c]); }
    float s = 0.f;
#pragma unroll
    for (int c = 0; c < CLOG; ++c) { vals[c] = __expf(vals[c] - mx); s += vals[c]; }
    const float inv = 1.f / s;
    float* pp = prob + (size_t)b * CLOG * HH * WW + hw;
#pragma unroll
    for (int c = 0; c < CLOG; ++c) pp[(size_t)c * HH * WW] = vals[c] * inv;
}

// ---------------------------------------------------------------------------
// Kernel 3: per-pixel cross terms, sim from Gram diagonals, stable top-k,
//           masked accumulation -> per-block partials (deterministic tree).
// ---------------------------------------------------------------------------
__global__ __launch_bounds__(256)
void loss_kernel(const float* __restrict__ prob, const float* __restrict__ gd,
                 const float* __restrict__ xe, float* __restrict__ partials) {
    __shared__ float red0[256];
    __shared__ float red1[256];
    __shared__ float red2[256];

    const int tid = threadIdx.x;
    const int idx = blockIdx.x * 256 + tid;              // exactly 32768 threads
    const int b   = idx >> 14;
    const int hw  = idx & 16383;
    const int h   = hw >> 7;
    const int w   = hw & 127;
    const int yl  = h >> 2;
    const int xl  = w >> 2;

    // center prob
    const float* pc = prob + (size_t)b * CLOG * HH * WW + hw;
    float p[CLOG];
    float psum = 0.f;
#pragma unroll
    for (int c = 0; c < CLOG; ++c) { p[c] = pc[(size_t)c * HH * WW]; psum += p[c]; }

    const float nc = gd[((((size_t)b * HL + yl) * 3 + 1) * 3 + 1) * WL + xl];

    float sim[9], cp[9], cn[9];
#pragma unroll
    for (int k = 0; k < 9; ++k) {
        const int dy = (k / 3 - 1) * 2;
        const int dx = (k % 3 - 1) * 2;
        const int h2 = h + dy;
        const int w2 = w + dx;
        const bool inb = (h2 >= 0) && (h2 < HH) && (w2 >= 0) && (w2 < WW);
        float cpos = 0.f, qs = 0.f, s;
        if (inb) {
            const float* qn = prob + (size_t)b * CLOG * HH * WW + (size_t)h2 * WW + w2;
#pragma unroll
            for (int c = 0; c < CLOG; ++c) {
                float qv = qn[(size_t)c * HH * WW];
                cpos += p[c] * qv;
                qs   += qv;
            }
            const int yl2 = h2 >> 2, xl2 = w2 >> 2;
            const float nn = gd[((((size_t)b * HL + yl2) * 3 + 1) * 3 + 1) * WL + xl2];
            const float g  = gd[((((size_t)b * HL + yl) * 3 + (yl2 - yl + 1)) * 3
                                 + (xl2 - xl + 1)) * WL + xl];
            const float dist = nc + nn - 2.f * g;
            s = __expf(-dist * 0.25f);
        } else {
            s = __expf(-nc * 0.25f);      // zero-padded neighbor: dist = ||f_c||^2
        }
        sim[k] = s;
        cp[k]  = cpos;
        cn[k]  = psum * qs - cpos;
    }

    // top-5 largest (stable: strict > keeps lowest index on ties, like lax.top_k)
    unsigned usedA = 0;
    float pos_acc = 0.f;
#pragma unroll
    for (int t = 0; t < 5; ++t) {
        float bv = -3.0e38f; int bi = 0;
#pragma unroll
        for (int k = 0; k < 9; ++k) {
            bool avail = ((usedA >> k) & 1u) == 0u;
            if (avail && sim[k] > bv) { bv = sim[k]; bi = k; }
        }
        usedA |= (1u << bi);
        float g = 0.f;
#pragma unroll
        for (int k = 0; k < 9; ++k) if (k == bi) g = cp[k];
        pos_acc += bv * (-g);
    }

    // top-4 smallest (strict < keeps lowest index on ties, like top_k(-sim))
    unsigned usedB = 0;
    float neg_acc = 0.f;
#pragma unroll
    for (int t = 0; t < 4; ++t) {
        float bv = 3.0e38f; int bi = 0;
#pragma unroll
        for (int k = 0; k < 9; ++k) {
            bool avail = ((usedB >> k) & 1u) == 0u;
            if (avail && sim[k] < bv) { bv = sim[k]; bi = k; }
        }
        usedB |= (1u << bi);
        float g = 0.f;
#pragma unroll
        for (int k = 0; k < 9; ++k) if (k == bi) g = cn[k];
        neg_acc += (1.f - bv) * (-g);
    }

    // mask = feats[:,0,:,:] > 0  ->  x_ema[b,0,yl,xl] > 0
    const bool msk = xe[(size_t)b * CF * HL * WL + (size_t)yl * WL + xl] > 0.f;
    red0[tid] = msk ? pos_acc : 0.f;
    red1[tid] = msk ? neg_acc : 0.f;
    red2[tid] = msk ? 1.f : 0.f;
    __syncthreads();

    for (int s2 = 128; s2 > 0; s2 >>= 1) {
        if (tid < s2) {
            red0[tid] += red0[tid + s2];
            red1[tid] += red1[tid + s2];
            red2[tid] += red2[tid + s2];
        }
        __syncthreads();
    }
    if (tid == 0) {
        partials[blockIdx.x * 3 + 0] = red0[0];
        partials[blockIdx.x * 3 + 1] = red1[0];
        partials[blockIdx.x * 3 + 2] = red2[0];
    }
}

// ---------------------------------------------------------------------------
// Kernel 4: deterministic fixed-order final reduction
// ---------------------------------------------------------------------------
__global__ __launch_bounds__(32)
void final_kernel(const float* __restrict__ partials, int nblocks,
                  float* __restrict__ out) {
    if (threadIdx.x == 0 && blockIdx.x == 0) {
        float sp = 0.f, sn = 0.f, cm = 0.f;
        for (int i = 0; i < nblocks; ++i) {
            sp += partials[i * 3 + 0];
            sn += partials[i * 3 + 1];
            cm += partials[i * 3 + 2];
        }
        out[0] = sp / (cm * 5.f);   // W_POS = 1, TOP_K+1 = 5
        out[1] = sn / (cm * 4.f);   // W_NEG = 1, TOP_K   = 4
    }
}

extern "C" void kernel_launch(void* const* d_in, const int* in_sizes, int n_in,
                              void* d_out, int out_size, void* d_ws, size_t ws_size,
                              hipStream_t stream) {
    const float* logits = (const float*)d_in[0];   // (2,19,128,128) f32
    const float* x_ema  = (const float*)d_in[1];   // (2,320,32,32)  f32
    // d_in[2] (img_trg) is unused by the reference.

    float* ws   = (float*)d_ws;
    float* gd   = ws + GD_OFF;
    float* prob = ws + PROB_OFF;
    float* part = ws + PART_OFF;

    dim3 g1(4, 96, BATCH);                 // tile-pair, (dy+1)*32+y, batch
    gram_wmma_kernel<<<g1, 32, 0, stream>>>(x_ema, gd);
    softmax_kernel  <<<128, 256, 0, stream>>>(logits, prob);
    loss_kernel     <<<128, 256, 0, stream>>>(prob, gd, x_ema, part);
    final_kernel    <<<1, 32, 0, stream>>>(part, 128, (float*)d_out);
}